// Capstone_32770600468617
// MI455X (gfx1250) — compile-verified
//
#include <hip/hip_runtime.h>
#include <hip/hip_bf16.h>
#include <math.h>

// ---------------------------------------------------------------------------
// CDNA5 (gfx1250, wave32) implementation of the hetero-GraphSAGE vote model.
// All dense GEMMs go through V_WMMA_F32_16X16X4_F32 (full fp32 precision,
// matrix pipe engaged). Edge features are recomputed in the scatter kernel.
// ---------------------------------------------------------------------------

typedef __attribute__((ext_vector_type(2))) float v2f;
typedef __attribute__((ext_vector_type(8))) float v8f;

#define DD   128
#define NB_   50000
#define NBV_  100000
#define NLT_  10000
#define NC_   2000
#define NT_   115
#define EV_   400000
#define ER_   50000
#define NVOTE_ 100000

// ---------------------------------------------------------------------------
// Generic WMMA fp32 GEMM:
//   C[M x N] (+)= (A[rowidx ? rowidx[m] : m, :K] * 1/max(rowcnt[m],1)) @ W + bias
// flags: bit0 = accumulate into C, bit1 = ReLU.
// Block = 128 threads (4 waves). Each wave owns one 16-col tile and FOUR
// 16-row tiles (64 rows), so each B fragment feeds 4 WMMAs (1.5 vmem
// issues per wmma). Grid = (ceil(M/64), N/64); N multiple of 64.
// Hot loop has NO predication: row indices are clamped (bogus rows discarded
// at store time); K remainder (K%4) handled once after the loop.
// Segment-mean scaling is applied to the accumulator in the epilogue.
// ---------------------------------------------------------------------------
__global__ void gemm_wmma_f32(const float* __restrict__ A,
                              const int*   __restrict__ rowidx,
                              const float* __restrict__ rowcnt,
                              const float* __restrict__ W,
                              const float* __restrict__ bias,
                              float* __restrict__ C,
                              int M, int K, int N, int lda, int flags)
{
    const int lane  = threadIdx.x & 31;
    const int wave  = threadIdx.x >> 5;
    const int l15   = lane & 15;
    const int khalf = lane >> 4;               // 0/1 -> k pair {0,1} vs {2,3}
    const int mt    = blockIdx.x;              // 64-row tile
    const int ct    = blockIdx.y * 4 + wave;   // 16-col tile
    const int ncol  = ct * 16 + l15;

    const float* Ap[4];
#pragma unroll
    for (int i = 0; i < 4; ++i) {
        const int mrow = mt * 64 + i * 16 + l15;
        const int mc = (mrow < M) ? mrow : (M - 1);   // clamp: always-safe loads
        const int ar = rowidx ? rowidx[mc] : mc;
        Ap[i] = A + (size_t)ar * lda + khalf * 2;
    }
    const float* Wp = W + (size_t)(khalf * 2) * N + ncol;

    v8f acc[4];
#pragma unroll
    for (int i = 0; i < 4; ++i)
        acc[i] = (v8f){0.f, 0.f, 0.f, 0.f, 0.f, 0.f, 0.f, 0.f};

    const int K4 = K & ~3;
#pragma unroll 2
    for (int k0 = 0; k0 < K4; k0 += 4) {
        v2f bf;
        bf.x = Wp[(size_t)k0 * N];
        bf.y = Wp[(size_t)k0 * N + N];
        v2f af[4];
#pragma unroll
        for (int i = 0; i < 4; ++i) {           // each pair merges into b64 load
            af[i].x = Ap[i][k0];
            af[i].y = Ap[i][k0 + 1];
        }
#pragma unroll
        for (int i = 0; i < 4; ++i)
            acc[i] = __builtin_amdgcn_wmma_f32_16x16x4_f32(
                false, af[i], false, bf, (short)0, acc[i], false, false);
    }
    if (K4 < K) {   // K%4 in {1,2,3}; runs once
        const int ka = K4 + khalf * 2;
        v2f bf = {0.f, 0.f};
        v2f af[4];
#pragma unroll
        for (int i = 0; i < 4; ++i) af[i] = (v2f){0.f, 0.f};
        if (ka < K) {
            bf.x = Wp[(size_t)K4 * N];
#pragma unroll
            for (int i = 0; i < 4; ++i) af[i].x = Ap[i][K4];
        }
        if (ka + 1 < K) {
            bf.y = Wp[(size_t)K4 * N + N];
#pragma unroll
            for (int i = 0; i < 4; ++i) af[i].y = Ap[i][K4 + 1];
        }
#pragma unroll
        for (int i = 0; i < 4; ++i)
            acc[i] = __builtin_amdgcn_wmma_f32_16x16x4_f32(
                false, af[i], false, bf, (short)0, acc[i], false, false);
    }

    const float bv = bias ? bias[ncol] : 0.0f;

#pragma unroll
    for (int i = 0; i < 4; ++i) {
        const int mbase = mt * 64 + i * 16 + khalf * 8;
        if (mbase + 8 <= M) {                  // fast path: whole half-tile valid
#pragma unroll
            for (int r = 0; r < 8; ++r) {
                const int m = mbase + r;
                float rsr = 1.0f;
                if (rowcnt) rsr = 1.0f / fmaxf(rowcnt[m], 1.0f);
                const size_t off = (size_t)m * N + ncol;
                float v = acc[i][r] * rsr + bv;
                if (flags & 1) v += C[off];
                if (flags & 2) v = fmaxf(v, 0.0f);
                C[off] = v;
            }
        } else {
#pragma unroll
            for (int r = 0; r < 8; ++r) {
                const int m = mbase + r;
                if (m < M) {
                    float rsr = 1.0f;
                    if (rowcnt) rsr = 1.0f / fmaxf(rowcnt[m], 1.0f);
                    const size_t off = (size_t)m * N + ncol;
                    float v = acc[i][r] * rsr + bv;
                    if (flags & 1) v += C[off];
                    if (flags & 2) v = fmaxf(v, 0.0f);
                    C[off] = v;
                }
            }
        }
    }
}

// ---------------------------------------------------------------------------
// Per-relation message scatter:  agg[di[e]] += h[si[e]] + t2v(t[e]) @ eW + eb
// One wave (32 lanes) per edge; 4 dims per lane; f32 global atomics (L2).
// ---------------------------------------------------------------------------
__global__ void sage_aggregate(const float* __restrict__ hsrc,
                               const int*   __restrict__ si,
                               const int*   __restrict__ di,
                               const float* __restrict__ tval,
                               int E,
                               const float* __restrict__ w0,
                               const float* __restrict__ b0,
                               const float* __restrict__ wk,
                               const float* __restrict__ bk,
                               const float* __restrict__ eW,  // 9 x 128
                               const float* __restrict__ eb,  // 128
                               float* __restrict__ agg,
                               float* __restrict__ cnt)
{
    const int e = blockIdx.x * 8 + (threadIdx.x >> 5);
    if (e >= E) return;
    const int lane = threadIdx.x & 31;
    const int s = si[e], d = di[e];
    const float t = tval[e];

    float tv[9];
    tv[0] = t * w0[0] + b0[0];
#pragma unroll
    for (int j = 0; j < 8; ++j) tv[1 + j] = sinf(t * wk[j] + bk[j]);

    const int c = lane * 4;
    const float* hp = hsrc + (size_t)s * DD + c;
    float m0 = eb[c + 0], m1 = eb[c + 1], m2 = eb[c + 2], m3 = eb[c + 3];
#pragma unroll
    for (int k = 0; k < 9; ++k) {
        const float* wp = eW + k * DD + c;
        m0 += tv[k] * wp[0]; m1 += tv[k] * wp[1];
        m2 += tv[k] * wp[2]; m3 += tv[k] * wp[3];
    }
    m0 += hp[0]; m1 += hp[1]; m2 += hp[2]; m3 += hp[3];

    float* ap = agg + (size_t)d * DD + c;
    atomicAdd(ap + 0, m0); atomicAdd(ap + 1, m1);
    atomicAdd(ap + 2, m2); atomicAdd(ap + 3, m3);
    if (lane == 0) atomicAdd(cnt + d, 1.0f);
}

// Plain row scatter-add (bill aggregation).
__global__ void scatter_add_rows(const float* __restrict__ hsrc,
                                 const int* __restrict__ si,
                                 const int* __restrict__ di, int E,
                                 float* __restrict__ agg, float* __restrict__ cnt)
{
    const int e = blockIdx.x * 8 + (threadIdx.x >> 5);
    if (e >= E) return;
    const int lane = threadIdx.x & 31;
    const int s = si[e], d = di[e];
    const int c = lane * 4;
    const float* hp = hsrc + (size_t)s * DD + c;
    float* ap = agg + (size_t)d * DD + c;
    atomicAdd(ap + 0, hp[0]); atomicAdd(ap + 1, hp[1]);
    atomicAdd(ap + 2, hp[2]); atomicAdd(ap + 3, hp[3]);
    if (lane == 0) atomicAdd(cnt + d, 1.0f);
}

// In-place LayerNorm over rows of width 128. One wave32 per row.
__global__ void layernorm_inplace(float* __restrict__ x,
                                  const float* __restrict__ g,
                                  const float* __restrict__ b, int M)
{
    const int row = blockIdx.x;
    if (row >= M) return;
    const int lane = threadIdx.x;
    const int c = lane * 4;
    float* xp = x + (size_t)row * DD + c;
    float v0 = xp[0], v1 = xp[1], v2 = xp[2], v3 = xp[3];

    float s = v0 + v1 + v2 + v3;
    for (int m = 16; m >= 1; m >>= 1) s += __shfl_xor(s, m, 32);
    const float mean = s * (1.0f / DD);

    const float d0 = v0 - mean, d1 = v1 - mean, d2 = v2 - mean, d3 = v3 - mean;
    float q = d0 * d0 + d1 * d1 + d2 * d2 + d3 * d3;
    for (int m = 16; m >= 1; m >>= 1) q += __shfl_xor(q, m, 32);
    const float inv = rsqrtf(q * (1.0f / DD) + 1e-5f);

    xp[0] = d0 * inv * g[c + 0] + b[c + 0];
    xp[1] = d1 * inv * g[c + 1] + b[c + 1];
    xp[2] = d2 * inv * g[c + 2] + b[c + 2];
    xp[3] = d3 * inv * g[c + 3] + b[c + 3];
}

// h_bill = 0.5*h0 + 0.5*agg/(cnt + 1e-6)
__global__ void combine_bill(const float* __restrict__ h0,
                             const float* __restrict__ bagg,
                             const float* __restrict__ bcnt,
                             float* __restrict__ hbill, int M)
{
    const int i = blockIdx.x * blockDim.x + threadIdx.x;
    if (i >= M * DD) return;
    const float c = bcnt[i >> 7] + 1e-6f;
    hbill[i] = 0.5f * h0[i] + 0.5f * (bagg[i] / c);
}

// b = ei_isv1[vote_bv], t = t_for_bill[b]
__global__ void vote_idx(const int* __restrict__ ei1, const int* __restrict__ vbv,
                         const int* __restrict__ tfb, int n,
                         int* __restrict__ bidx, int* __restrict__ tidx)
{
    const int i = blockIdx.x * blockDim.x + threadIdx.x;
    if (i >= n) return;
    const int b = ei1[vbv[i]];
    bidx[i] = b;
    tidx[i] = tfb[b];
}

// Per-vote attention: softmax over H=4 heads of dot(Q_h,K_h)/sqrt(32); ctxt=a*V.
__global__ void vote_attn(const float* __restrict__ Q, const float* __restrict__ Kb,
                          const float* __restrict__ V, float* __restrict__ ctxt, int n)
{
    const int e = blockIdx.x * 8 + (threadIdx.x >> 5);
    if (e >= n) return;
    const int lane = threadIdx.x & 31;
    const size_t base = (size_t)e * DD + lane * 4;

    const float q0 = Q[base + 0], q1 = Q[base + 1], q2 = Q[base + 2], q3 = Q[base + 3];
    const float k0 = Kb[base + 0], k1 = Kb[base + 1], k2 = Kb[base + 2], k3 = Kb[base + 3];
    float p = q0 * k0 + q1 * k1 + q2 * k2 + q3 * k3;
    // reduce within 8-lane head group (dk = 32 -> 8 lanes * 4)
    p += __shfl_xor(p, 1, 32);
    p += __shfl_xor(p, 2, 32);
    p += __shfl_xor(p, 4, 32);
    const float s = p * 0.17677669529663687f;  // 1/sqrt(32)

    const float s0 = __shfl(s, 0, 32);
    const float s1 = __shfl(s, 8, 32);
    const float s2 = __shfl(s, 16, 32);
    const float s3 = __shfl(s, 24, 32);
    const float mx = fmaxf(fmaxf(s0, s1), fmaxf(s2, s3));
    const float e0 = expf(s0 - mx), e1 = expf(s1 - mx);
    const float e2 = expf(s2 - mx), e3 = expf(s3 - mx);
    const float den = e0 + e1 + e2 + e3;
    const int h = lane >> 3;
    const float a = ((h == 0) ? e0 : (h == 1) ? e1 : (h == 2) ? e2 : e3) / den;

    ctxt[base + 0] = a * V[base + 0];
    ctxt[base + 1] = a * V[base + 1];
    ctxt[base + 2] = a * V[base + 2];
    ctxt[base + 3] = a * V[base + 3];
}

// z = concat(h_lt[vote_lt], h_bv[vote_bv], h_top[tidx], ctxt)  (N x 512)
__global__ void build_z(const float* __restrict__ h_lt, const int* __restrict__ vlt,
                        const float* __restrict__ h_bv, const int* __restrict__ vbv,
                        const float* __restrict__ h_top, const int* __restrict__ tidx,
                        const float* __restrict__ ctxt, float* __restrict__ z, int n)
{
    const int e = blockIdx.x * 8 + (threadIdx.x >> 5);
    if (e >= n) return;
    const int lane = threadIdx.x & 31;
    const int c = lane * 4;
    float* zp = z + (size_t)e * 512;
    const float* a = h_lt + (size_t)vlt[e] * DD + c;
    const float* b = h_bv + (size_t)vbv[e] * DD + c;
    const float* t = h_top + (size_t)tidx[e] * DD + c;
    const float* x = ctxt + (size_t)e * DD + c;
#pragma unroll
    for (int j = 0; j < 4; ++j) {
        zp[c + j]       = a[j];
        zp[128 + c + j] = b[j];
        zp[256 + c + j] = t[j];
        zp[384 + c + j] = x[j];
    }
}

// out[e, 0..2] = t2[e, :] @ v_Wo (128x3) + v_bo
__global__ void final_head(const float* __restrict__ t2, const float* __restrict__ Wo,
                           const float* __restrict__ bo, float* __restrict__ out, int n)
{
    const int e = blockIdx.x * 8 + (threadIdx.x >> 5);
    if (e >= n) return;
    const int lane = threadIdx.x & 31;
    const int c = lane * 4;
    const float* xp = t2 + (size_t)e * DD + c;
    const float x0 = xp[0], x1 = xp[1], x2 = xp[2], x3 = xp[3];
    float p[3];
#pragma unroll
    for (int j = 0; j < 3; ++j) {
        p[j] = x0 * Wo[(c + 0) * 3 + j] + x1 * Wo[(c + 1) * 3 + j] +
               x2 * Wo[(c + 2) * 3 + j] + x3 * Wo[(c + 3) * 3 + j];
        for (int m = 16; m >= 1; m >>= 1) p[j] += __shfl_xor(p[j], m, 32);
    }
    if (lane == 0) {
        out[(size_t)e * 3 + 0] = p[0] + bo[0];
        out[(size_t)e * 3 + 1] = p[1] + bo[1];
        out[(size_t)e * 3 + 2] = p[2] + bo[2];
    }
}

// ---------------------------------------------------------------------------
extern "C" void kernel_launch(void* const* d_in, const int* in_sizes, int n_in,
                              void* d_out, int out_size, void* d_ws, size_t ws_size,
                              hipStream_t stream)
{
    (void)in_sizes; (void)n_in; (void)out_size; (void)ws_size;

    const float* x_bill = (const float*)d_in[0];
    const float* x_bv   = (const float*)d_in[1];
    const float* x_lt   = (const float*)d_in[2];
    const float* x_comm = (const float*)d_in[3];
    const float* pb_W   = (const float*)d_in[4];
    const float* pb_b   = (const float*)d_in[5];
    const float* pbv_W  = (const float*)d_in[6];
    const float* pbv_b  = (const float*)d_in[7];
    const float* plt_W1 = (const float*)d_in[8];
    const float* plt_b1 = (const float*)d_in[9];
    const float* plt_W2 = (const float*)d_in[10];
    const float* plt_b2 = (const float*)d_in[11];
    const float* pc_W1  = (const float*)d_in[12];
    const float* pc_b1  = (const float*)d_in[13];
    const float* pc_W2  = (const float*)d_in[14];
    const float* pc_b2  = (const float*)d_in[15];
    const float* topic  = (const float*)d_in[16];
    const float* t2v_w0 = (const float*)d_in[17];
    const float* t2v_b0 = (const float*)d_in[18];
    const float* t2v_wk = (const float*)d_in[19];
    const float* t2v_bk = (const float*)d_in[20];
    const float* edge_W = (const float*)d_in[21];
    const float* edge_b = (const float*)d_in[22];
    const float* sage_Wl = (const float*)d_in[23];
    const float* sage_bl = (const float*)d_in[24];
    const float* sage_Wr = (const float*)d_in[25];
    const float* sage_br = (const float*)d_in[26];
    const float* ln_g   = (const float*)d_in[27];
    const float* ln_b   = (const float*)d_in[28];
    const float* xq_W = (const float*)d_in[29];
    const float* xq_b = (const float*)d_in[30];
    const float* xk_W = (const float*)d_in[31];
    const float* xk_b = (const float*)d_in[32];
    const float* xv_W = (const float*)d_in[33];
    const float* xv_b = (const float*)d_in[34];
    const float* xo_W = (const float*)d_in[35];
    const float* xo_b = (const float*)d_in[36];
    const float* v_W1 = (const float*)d_in[37];
    const float* v_b1 = (const float*)d_in[38];
    const float* v_W2 = (const float*)d_in[39];
    const float* v_b2 = (const float*)d_in[40];
    const float* v_Wo = (const float*)d_in[41];
    const float* v_bo = (const float*)d_in[42];
    const float* t_isv   = (const float*)d_in[43];
    const float* t_vote  = (const float*)d_in[44];
    const float* t_rev   = (const float*)d_in[45];
    const float* t_about = (const float*)d_in[46];
    const int* ei_isv   = (const int*)d_in[47];
    const int* ei_vote  = (const int*)d_in[48];
    const int* ei_rev   = (const int*)d_in[49];
    const int* ei_about = (const int*)d_in[50];
    const int* t_for_bill = (const int*)d_in[51];
    const int* vote_lt  = (const int*)d_in[52];
    const int* vote_bv  = (const int*)d_in[53];

    static const int SZ[5] = {NB_, NBV_, NLT_, NC_, NT_};

    // ---- workspace carve-up (all 256B aligned) ----
    size_t off = 0;
    auto alloc = [&](size_t bytes) -> char* {
        char* p = (char*)d_ws + off;
        off = (off + bytes + 255) & ~(size_t)255;
        return p;
    };
    size_t setBytes = 0;
    for (int t = 0; t < 5; ++t) setBytes += (size_t)SZ[t] * DD * 4;

    char* hA_base = alloc(setBytes);
    char* hB_base = alloc(setBytes);
    float* hA[5]; float* hB[5];
    {
        size_t o = 0;
        for (int t = 0; t < 5; ++t) {
            hA[t] = (float*)(hA_base + o);
            hB[t] = (float*)(hB_base + o);
            o += (size_t)SZ[t] * DD * 4;
        }
    }
    float* agg    = (float*)alloc((size_t)NBV_ * DD * 4);
    float* cnt    = (float*)alloc((size_t)NBV_ * 4);
    float* hid_lt = (float*)alloc((size_t)NLT_ * 64 * 4);
    float* hid_cm = (float*)alloc((size_t)NC_ * DD * 4);
    float* bagg   = (float*)alloc((size_t)NB_ * DD * 4);
    float* bcnt   = (float*)alloc((size_t)NB_ * 4);
    float* hbill  = (float*)alloc((size_t)NB_ * DD * 4);
    int*   bidx   = (int*)alloc((size_t)NVOTE_ * 4);
    int*   tidx   = (int*)alloc((size_t)NVOTE_ * 4);
    char*  vcr    = alloc((size_t)2 * NVOTE_ * DD * 4);  // V | ctxt_raw ; later t1
    float* zbuf   = (float*)alloc((size_t)NVOTE_ * 512 * 4);

    float* Qb    = (float*)hA_base;                       // reuse set A after GNN
    float* Kb    = agg;                                   // reuse agg after GNN
    float* Vb    = (float*)vcr;
    float* Craw  = (float*)(vcr + (size_t)NVOTE_ * DD * 4);
    float* Cproj = agg;                                   // overwrite K after attention
    float* t1buf = (float*)vcr;                           // reuse V|Craw (102 MB)
    float* t2buf = (float*)hA_base;                       // reuse Q after z is built

    auto gemm = [&](const float* A, const int* ridx, const float* rcnt,
                    const float* W, const float* bias, float* C,
                    int M, int K, int N, int lda, int flags) {
        dim3 grid((M + 63) / 64, N / 64);
        gemm_wmma_f32<<<grid, dim3(128), 0, stream>>>(A, ridx, rcnt, W, bias, C,
                                                      M, K, N, lda, flags);
    };

    // ---- 1) node encoders -> set A ----
    gemm(x_bill, nullptr, nullptr, pb_W, pb_b, hA[0], NB_, 770, 128, 770, 0);
    gemm(x_bv,   nullptr, nullptr, pbv_W, pbv_b, hA[1], NBV_, 390, 128, 390, 0);
    gemm(x_lt,   nullptr, nullptr, plt_W1, plt_b1, hid_lt, NLT_, 4, 64, 4, 2);
    gemm(hid_lt, nullptr, nullptr, plt_W2, plt_b2, hA[2], NLT_, 64, 128, 64, 0);
    gemm(x_comm, nullptr, nullptr, pc_W1, pc_b1, hid_cm, NC_, 65, 128, 65, 2);
    gemm(hid_cm, nullptr, nullptr, pc_W2, pc_b2, hA[3], NC_, 128, 128, 128, 0);
    hipMemcpyAsync(hA[4], topic, (size_t)NT_ * DD * 4, hipMemcpyDeviceToDevice, stream);

    // ---- 2) relations ----
    struct Rel { int s, d; const int* si; const int* di; const float* tv; int E; };
    const Rel rels[8] = {
        {1, 0, ei_isv,        ei_isv + NBV_,  t_isv,   NBV_},
        {0, 1, ei_isv + NBV_, ei_isv,         t_isv,   NBV_},
        {2, 1, ei_vote,       ei_vote + EV_,  t_vote,  EV_},
        {1, 2, ei_vote + EV_, ei_vote,        t_vote,  EV_},
        {3, 0, ei_rev,        ei_rev + ER_,   t_rev,   ER_},
        {0, 3, ei_rev + ER_,  ei_rev,         t_rev,   ER_},
        {0, 4, ei_about,      ei_about + NB_, t_about, NB_},
        {4, 0, ei_about + NB_,ei_about,       t_about, NB_},
    };

    float* cur[5]; float* nxt[5];
    char* curBase = hA_base; char* nxtBase = hB_base;
    for (int t = 0; t < 5; ++t) { cur[t] = hA[t]; nxt[t] = hB[t]; }

    // ---- 3) SAGE layers ----
    for (int l = 0; l < 3; ++l) {
        hipMemsetAsync(nxtBase, 0, setBytes, stream);
        for (int r = 0; r < 8; ++r) {
            const Rel& R = rels[r];
            const int Md = SZ[R.d];
            hipMemsetAsync(agg, 0, (size_t)Md * DD * 4, stream);
            hipMemsetAsync(cnt, 0, (size_t)Md * 4, stream);
            sage_aggregate<<<dim3((R.E + 7) / 8), dim3(256), 0, stream>>>(
                cur[R.s], R.si, R.di, R.tv, R.E,
                t2v_w0, t2v_b0, t2v_wk, t2v_bk,
                edge_W + (size_t)r * 9 * DD, edge_b + (size_t)r * DD,
                agg, cnt);
            const size_t wo = ((size_t)l * 8 + r) * DD * DD;
            const size_t bo2 = ((size_t)l * 8 + r) * DD;
            gemm(agg,      nullptr, cnt,     sage_Wl + wo, sage_bl + bo2, nxt[R.d],
                 Md, 128, 128, 128, 1);
            gemm(cur[R.d], nullptr, nullptr, sage_Wr + wo, sage_br + bo2, nxt[R.d],
                 Md, 128, 128, 128, 1);
        }
        for (int t = 0; t < 5; ++t)
            layernorm_inplace<<<dim3(SZ[t]), dim3(32), 0, stream>>>(
                nxt[t], ln_g + ((size_t)l * 5 + t) * DD, ln_b + ((size_t)l * 5 + t) * DD,
                SZ[t]);
        for (int t = 0; t < 5; ++t) { float* tmp = cur[t]; cur[t] = nxt[t]; nxt[t] = tmp; }
        char* tb = curBase; curBase = nxtBase; nxtBase = tb;
    }
    // final h lives in set B (3 swaps); set A + agg are now reusable scratch.

    // ---- 4) bill aggregation ----
    hipMemsetAsync(bagg, 0, (size_t)NB_ * DD * 4, stream);
    hipMemsetAsync(bcnt, 0, (size_t)NB_ * 4, stream);
    scatter_add_rows<<<dim3((NBV_ + 7) / 8), dim3(256), 0, stream>>>(
        cur[1], ei_isv, ei_isv + NBV_, NBV_, bagg, bcnt);
    combine_bill<<<dim3((NB_ * DD + 255) / 256), dim3(256), 0, stream>>>(
        cur[0], bagg, bcnt, hbill, NB_);

    // ---- 5) vote head ----
    vote_idx<<<dim3((NVOTE_ + 255) / 256), dim3(256), 0, stream>>>(
        ei_isv + NBV_, vote_bv, t_for_bill, NVOTE_, bidx, tidx);

    gemm(cur[2], vote_lt, nullptr, xq_W, xq_b, Qb, NVOTE_, 128, 128, 128, 0);
    gemm(hbill,  bidx,    nullptr, xk_W, xk_b, Kb, NVOTE_, 128, 128, 128, 0);
    gemm(hbill,  bidx,    nullptr, xv_W, xv_b, Vb, NVOTE_, 128, 128, 128, 0);

    vote_attn<<<dim3((NVOTE_ + 7) / 8), dim3(256), 0, stream>>>(Qb, Kb, Vb, Craw, NVOTE_);
    gemm(Craw, nullptr, nullptr, xo_W, xo_b, Cproj, NVOTE_, 128, 128, 128, 0);

    build_z<<<dim3((NVOTE_ + 7) / 8), dim3(256), 0, stream>>>(
        cur[2], vote_lt, cur[1], vote_bv, cur[4], tidx, Cproj, zbuf, NVOTE_);

    gemm(zbuf,  nullptr, nullptr, v_W1, v_b1, t1buf, NVOTE_, 512, 256, 512, 2);
    gemm(t1buf, nullptr, nullptr, v_W2, v_b2, t2buf, NVOTE_, 256, 128, 256, 0);

    final_head<<<dim3((NVOTE_ + 7) / 8), dim3(256), 0, stream>>>(
        t2buf, v_Wo, v_bo, (float*)d_out, NVOTE_);
}